// NeuralODE_71863392796883
// MI455X (gfx1250) — compile-verified
//
#include <hip/hip_runtime.h>

typedef __attribute__((ext_vector_type(2))) float v2f;
typedef __attribute__((ext_vector_type(8))) float v8f;

// ---------------------------------------------------------------------------
// Kernel 1: scalar prefix s[n] such that out[n,:] = y0 + s[n]*w.
// One workgroup of 1024 threads, 4 steps per thread, LDS Hillis-Steele scan.
// ---------------------------------------------------------------------------
__global__ void __launch_bounds__(1024)
tsit5_scalar_scan(const float* __restrict__ ts, float* __restrict__ s, int T) {
    // Tsit5 nodes and 5th-order weights (as in diffrax.Tsit5)
    constexpr float kC[7] = {0.0f, 0.161f, 0.327f, 0.9f,
                             0.9800255409045097f, 1.0f, 1.0f};
    constexpr float kB[7] = {0.09646076681806523f, 0.01f, 0.4798896504144996f,
                             1.379008574103742f, -3.290069515436081f,
                             2.324710524099774f, 0.0f};

    __shared__ float partial[1024];
    const int tid = threadIdx.x;

    float v[4];
    float local = 0.0f;
#pragma unroll
    for (int j = 0; j < 4; ++j) {
        const int n = tid * 4 + j;   // s-index; increment feeding s[n] is step n-1
        float inc = 0.0f;
        if (n >= 1 && n < T) {
            const float t  = ts[n - 1];
            const float dt = ts[n] - ts[n - 1];
            float acc = 0.0f;
#pragma unroll
            for (int st = 0; st < 7; ++st)
                acc += kB[st] * (t + kC[st] * dt);
            inc = dt * acc;
        }
        local += inc;
        v[j] = local;
    }
    partial[tid] = local;
    __syncthreads();

#pragma unroll 1
    for (int off = 1; off < 1024; off <<= 1) {
        const float add = (tid >= off) ? partial[tid - off] : 0.0f;
        __syncthreads();
        partial[tid] += add;
        __syncthreads();
    }
    const float excl = (tid > 0) ? partial[tid - 1] : 0.0f;

#pragma unroll
    for (int j = 0; j < 4; ++j) {
        const int n = tid * 4 + j;
        if (n < T) s[n] = excl + v[j];
    }
}

// ---------------------------------------------------------------------------
// Kernel 2 (fast path, D compile-time): out[m,n] = y0[n] + s[m]*w[n] via
// V_WMMA_F32_16X16X4_F32. Each wave owns a 16x64 strip = 4 N-tiles sharing
// one A matrix (s column). All 32 stores per wave use one 64-bit base plus
// compile-time IOFFSETs (max 7*D*4 + 48*4 < 2^23).
// Loads are unconditional (indices in-range for all lanes); the K=0 lane
// masking is done with selects so EXEC stays all-ones (no saveexec branches,
// and WMMA requires full EXEC anyway).
//   A (16x4, 2 VGPRs): column K=0 = s[row0+m], rest zero
//   B (4x16, 2 VGPRs): row    K=0 = w[col+n],  rest zero
//   C (8 VGPRs):       broadcast y0[col + lane%16]
// D layout: VGPR r, lane l -> M = r + 8*(l>=16), N = l%16.
// Grid: x = D/512 column strips (8 waves * 64 cols), y = T/16 row tiles.
// ---------------------------------------------------------------------------
template <int DD>
__global__ void __launch_bounds__(256)
ode_outer_wmma_fast(const float* __restrict__ s, const float* __restrict__ y0,
                    const float* __restrict__ w, float* __restrict__ out) {
    const int lane = threadIdx.x & 31;
    const int wave = threadIdx.x >> 5;
    const int row0 = blockIdx.y * 16;
    const int col0 = blockIdx.x * 512 + wave * 64;
    const int l15  = lane & 15;
    const bool lo  = (lane < 16);

    const float sv = s[row0 + l15];    // unconditional, in-range for all lanes
    v2f a;
    a.x = lo ? sv : 0.0f;              // K=0 column (lanes 0-15), cndmask
    a.y = 0.0f;                        // K=1..3 zero

    float* p = out + (size_t)(row0 + 8 * (lane >> 4)) * (size_t)DD + col0 + l15;

#pragma unroll
    for (int i = 0; i < 4; ++i) {
        const float wv = w[col0 + 16 * i + l15];   // unconditional load
        const float c0 = y0[col0 + 16 * i + l15];

        v2f b;
        b.x = lo ? wv : 0.0f;          // K=0 row (lanes 0-15), cndmask
        b.y = 0.0f;

        v8f c = {c0, c0, c0, c0, c0, c0, c0, c0};

        // 8 args: (neg_a, A, neg_b, B, c_mod, C, reuse_a, reuse_b)
        v8f d = __builtin_amdgcn_wmma_f32_16x16x4_f32(
            false, a, false, b, (short)0, c, false, false);

#pragma unroll
        for (int r = 0; r < 8; ++r)
            p[(size_t)r * DD + 16 * i] = d[r];     // constant IOFFSET per store
    }
}

// ---------------------------------------------------------------------------
// Generic fallback (any T, D): plain per-element FMA.
// ---------------------------------------------------------------------------
__global__ void __launch_bounds__(256)
ode_outer_generic(const float* __restrict__ s, const float* __restrict__ y0,
                  const float* __restrict__ w, float* __restrict__ out,
                  int D, long long total) {
    long long i = (long long)blockIdx.x * blockDim.x + threadIdx.x;
    long long stride = (long long)gridDim.x * blockDim.x;
    for (; i < total; i += stride) {
        const int d = (int)(i % D);
        const int t = (int)(i / D);
        out[i] = fmaf(s[t], w[d], y0[d]);
    }
}

// ---------------------------------------------------------------------------
// Launch: d_in[0]=ts (T), d_in[1]=y0 (D), d_in[2]=W (1xD). d_out = (T,D) f32.
// d_ws holds s[T].
// ---------------------------------------------------------------------------
extern "C" void kernel_launch(void* const* d_in, const int* in_sizes, int n_in,
                              void* d_out, int out_size, void* d_ws, size_t ws_size,
                              hipStream_t stream) {
    const float* ts = (const float*)d_in[0];
    const float* y0 = (const float*)d_in[1];
    const float* W  = (const float*)d_in[2];
    float* out = (float*)d_out;
    float* s   = (float*)d_ws;

    const int T = in_sizes[0];   // 4096
    const int D = in_sizes[1];   // 8192

    tsit5_scalar_scan<<<1, 1024, 0, stream>>>(ts, s, T);

    if (D == 8192 && (T % 16) == 0) {
        dim3 grid(D / 512, T / 16);   // (16, 256)
        ode_outer_wmma_fast<8192><<<grid, 256, 0, stream>>>(s, y0, W, out);
    } else {
        const long long total = (long long)T * D;
        int blocks = (int)((total + 1023) / 1024);
        if (blocks > 65535) blocks = 65535;
        ode_outer_generic<<<blocks, 256, 0, stream>>>(s, y0, W, out, D, total);
    }
}